// EHD_Layer_87892210745824
// MI455X (gfx1250) — compile-verified
//
#include <hip/hip_runtime.h>
#include <hip/hip_bf16.h>

// EHD layer, fully fused for MI455X (gfx1250).
// Memory-bound: ~597MB out + 64MB in => ~28us floor @ 23.3 TB/s.
// Conv (2.4 GFLOP) runs on the f32 WMMA pipe (V_WMMA_F32_16X16X4_F32).

typedef float v2f __attribute__((ext_vector_type(2)));
typedef float v8f __attribute__((ext_vector_type(8)));

#define OUTD   1018          // 1024 - 2 (conv) - 4 (pool)
#define TILE   28            // output tile per block
#define REG    32            // index-map region = TILE + 4
#define XR     34            // x region = REG + 2
#define XP     36            // padded LDS pitch for x
#define NB     9
#define THRESH 0.9f

__global__ __launch_bounds__(256)
void ehd_fused_kernel(const float* __restrict__ x,
                      const float* __restrict__ masks,
                      float* __restrict__ out)
{
    __shared__ float         xs[XR * XP];        // 4896 B  staged input tile
    __shared__ float         mlds[16 * 12];      //  768 B  filters padded to 16ch x 12 taps
    __shared__ unsigned char idxm[REG * REG];    // 1024 B  per-pixel bin index
    __shared__ float         chan[8 * 128];      // 4096 B  per-wave 16px x 8ch transpose
    __shared__ unsigned int  colLo[TILE * REG];  // 3584 B  packed vertical sums, bins 0-4
    __shared__ unsigned int  colHi[TILE * REG];  // 3584 B  packed vertical sums, bins 5-8

    const int tid   = threadIdx.x;
    const int wave  = tid >> 5;
    const int lane  = tid & 31;
    const int batch = blockIdx.z;
    const int tr0   = blockIdx.y * TILE;
    const int tc0   = blockIdx.x * TILE;

    const float* xb = x + (size_t)batch * 1024 * 1024;

    // ---- Phase 0: stage x tile (clamped at image edge) + padded filter matrix
    for (int i = tid; i < XR * XR; i += 256) {
        int r = i / XR, c = i % XR;
        int gr = tr0 + r; if (gr > 1023) gr = 1023;
        int gc = tc0 + c; if (gc > 1023) gc = 1023;
        xs[r * XP + c] = xb[(size_t)gr * 1024 + gc];
    }
    if (tid < 192) {
        int n = tid / 12, t = tid % 12;
        mlds[tid] = (n < 8 && t < 9) ? masks[n * 9 + t] : 0.0f;
    }
    __syncthreads();

    // ---- Phase 1: 3x3 conv (8 filters) via f32 WMMA, argmax -> bin index
    // Group = 16 consecutive pixels of one region row. 64 groups, 8 per wave.
    // WMMA f32 16x16x4 operand layout (wave32):
    //   A (16x4):  lane L: m=L&15, kh=L>>4; a[j] = A[m][2*kh + j]
    //   B (4x16):  lane L: n=L&15, kh=L>>4; b[j] = B[2*kh + j][n]
    //   D (16x16): lane L: n=L&15; d[v] = D[8*(L>>4) + v][n]
    const int mL = lane & 15;
    const int kh = lane >> 4;
    for (int g = wave; g < 64; g += 8) {
        const int r  = g >> 1;          // region row 0..31
        const int cb = (g & 1) << 4;    // region col base 0 or 16
        v8f acc = {0.f, 0.f, 0.f, 0.f, 0.f, 0.f, 0.f, 0.f};
        #pragma unroll
        for (int chk = 0; chk < 3; ++chk) {         // K = 12 (9 taps + 3 pad)
            const int t0 = chk * 4 + kh * 2;
            const int t1 = t0 + 1;
            v2f a, bm;
            a.x  = (t0 < 9) ? xs[(r + t0 / 3) * XP + cb + mL + (t0 % 3)] : 0.0f;
            a.y  = (t1 < 9) ? xs[(r + t1 / 3) * XP + cb + mL + (t1 % 3)] : 0.0f;
            bm.x = mlds[mL * 12 + t0];              // zero-padded for t>=9, n>=8
            bm.y = mlds[mL * 12 + t1];
            acc = __builtin_amdgcn_wmma_f32_16x16x4_f32(
                      false, a, false, bm, (short)0, acc, false, false);
        }
        // scatter D (pixel-major) into LDS so each lane can scan 8 channels
        if (mL < 8) {
            #pragma unroll
            for (int v = 0; v < 8; ++v)
                chan[wave * 128 + (kh * 8 + v) * 8 + mL] = acc[v];
        }
        asm volatile("s_wait_dscnt 0" ::: "memory");  // per-wave LDS RAW order
        const int p = lane & 15;                      // pixel within group
        float best = chan[wave * 128 + p * 8];
        int   bi   = 0;
        #pragma unroll
        for (int n2 = 1; n2 < 8; ++n2) {
            float v = chan[wave * 128 + p * 8 + n2];
            if (v > best) { best = v; bi = n2; }      // first-max = jnp.argmax
        }
        int bin = (best < THRESH) ? 8 : bi;
        if (lane < 16)
            idxm[r * REG + cb + p] = (unsigned char)bin;
        asm volatile("" ::: "memory");
    }
    __syncthreads();

    // ---- Phase 2: vertical 5-sum of packed one-hots (6-bit fields, max 5)
    for (int i = tid; i < TILE * REG; i += 256) {
        int r = i / REG, c = i % REG;
        unsigned lo = 0u, hi = 0u;
        #pragma unroll
        for (int dr = 0; dr < 5; ++dr) {
            int bv = idxm[(r + dr) * REG + c];
            lo += (bv < 5)  ? (1u << (6 * bv))        : 0u;
            hi += (bv >= 5) ? (1u << (6 * bv - 30))   : 0u;
        }
        colLo[i] = lo;
        colHi[i] = hi;
    }
    __syncthreads();

    // ---- Phase 3: horizontal 5-sum (fields max 25 < 64), scale, store
    const float scale = 1.0f / 25.0f;
    float* ob = out + (size_t)batch * NB * OUTD * OUTD;
    for (int i = tid; i < TILE * TILE; i += 256) {
        int r = i / TILE, c = i % TILE;
        int gr = tr0 + r, gc = tc0 + c;
        if (gr >= OUTD || gc >= OUTD) continue;
        unsigned hLo = 0u, hHi = 0u;
        #pragma unroll
        for (int dc = 0; dc < 5; ++dc) {
            hLo += colLo[r * REG + c + dc];
            hHi += colHi[r * REG + c + dc];
        }
        size_t base = (size_t)gr * OUTD + gc;
        #pragma unroll
        for (int bn = 0; bn < 5; ++bn)
            ob[(size_t)bn * OUTD * OUTD + base] =
                (float)((hLo >> (6 * bn)) & 63u) * scale;
        #pragma unroll
        for (int bn = 0; bn < 4; ++bn)
            ob[(size_t)(5 + bn) * OUTD * OUTD + base] =
                (float)((hHi >> (6 * bn)) & 63u) * scale;
    }
}

extern "C" void kernel_launch(void* const* d_in, const int* in_sizes, int n_in,
                              void* d_out, int out_size, void* d_ws, size_t ws_size,
                              hipStream_t stream) {
    (void)in_sizes; (void)n_in; (void)out_size; (void)d_ws; (void)ws_size;
    const float* x     = (const float*)d_in[0];   // [16,1,1024,1024] f32
    const float* masks = (const float*)d_in[1];   // [8,1,3,3] f32
    float*       out   = (float*)d_out;           // [16,9,1018,1018] f32

    const int nt = (OUTD + TILE - 1) / TILE;      // 37
    dim3 grid(nt, nt, 16);
    dim3 block(256);
    ehd_fused_kernel<<<grid, block, 0, stream>>>(x, masks, out);
}